// MultiHeadAttention_1279900254392
// MI455X (gfx1250) — compile-verified
//
#include <hip/hip_runtime.h>
#include <math.h>

#define BATCH 4
#define EDIM  128
#define SEQ   4096
#define KTILE 16

typedef __attribute__((ext_vector_type(2))) float v2f;
typedef __attribute__((ext_vector_type(8))) float v8f;

__device__ __forceinline__ v8f wmma_f32(v2f a, v2f b, v8f c) {
    // D = A(16x4, f32) * B(4x16, f32) + C(16x16, f32)
    return __builtin_amdgcn_wmma_f32_16x16x4_f32(
        /*neg_a=*/false, a, /*neg_b=*/false, b,
        /*c_mod=*/(short)0, c, /*reuse_a=*/false, /*reuse_b=*/false);
}

// ---------------------------------------------------------------------------
// Kernel 1: QKV projection.
//   Qt = Wq @ x[b] + bq   (M=E, N=S, K=E), same for K, V.  x[b] is already
//   [E,S] row-major so no transpose is needed anywhere.
// Writes: q_ws [B,S,E], k_ws [B,E,S], v_ws [B,S,E].
// Block = 128 threads (4 waves). Wave w computes a 16(M) x 64(N) tile for all
// three weight matrices simultaneously (shared B fragments from x).
// ---------------------------------------------------------------------------
__global__ __launch_bounds__(128) void qkv_proj_kernel(
    const float* __restrict__ x,
    const float* __restrict__ Wq, const float* __restrict__ bq,
    const float* __restrict__ Wk, const float* __restrict__ bk,
    const float* __restrict__ Wv, const float* __restrict__ bv,
    float* __restrict__ qws, float* __restrict__ kws, float* __restrict__ vws)
{
    const int lane = threadIdx.x & 31;
    const int wave = threadIdx.x >> 5;
    const int h    = lane >> 4;     // half-wave (selects K sub-pair / row group)
    const int ln   = lane & 15;

    const int n0 = blockIdx.x * 64;              // S-column tile base
    const int m0 = blockIdx.y * 64 + wave * 16;  // output-feature tile base
    const int b  = blockIdx.z;

    const float* xb = x + (size_t)b * EDIM * SEQ;

    v8f aq[4], ak[4], av[4];
#pragma unroll
    for (int j = 0; j < 4; ++j) { aq[j] = (v8f){}; ak[j] = (v8f){}; av[j] = (v8f){}; }

    const int arow = m0 + ln;      // A-fragment row for this lane
    const int ksub = h * 2;        // K offset within the 4-wide K step

#pragma unroll 4
    for (int k0 = 0; k0 < EDIM; k0 += 4) {
        const int kb = k0 + ksub;
        // A fragments (weights, row-major [E,E])
        v2f aQ = *(const v2f*)(Wq + arow * EDIM + kb);
        v2f aK = *(const v2f*)(Wk + arow * EDIM + kb);
        v2f aV = *(const v2f*)(Wv + arow * EDIM + kb);
#pragma unroll
        for (int j = 0; j < 4; ++j) {
            const int col = n0 + j * 16 + ln;
            v2f bF;
            bF.x = xb[(size_t)kb * SEQ + col];
            bF.y = xb[(size_t)(kb + 1) * SEQ + col];
            aq[j] = wmma_f32(aQ, bF, aq[j]);
            ak[j] = wmma_f32(aK, bF, ak[j]);
            av[j] = wmma_f32(aV, bF, av[j]);
        }
    }

    // biases: feature index i = m0 + r + 8*h  (C-layout row mapping)
    float biq[8], bik[8], biv[8];
#pragma unroll
    for (int r = 0; r < 8; ++r) {
        const int i = m0 + r + 8 * h;
        biq[r] = bq[i]; bik[r] = bk[i]; biv[r] = bv[i];
    }

#pragma unroll
    for (int j = 0; j < 4; ++j) {
        const int s = n0 + j * 16 + ln;
#pragma unroll
        for (int r = 0; r < 8; ++r) {
            const int i = m0 + r + 8 * h;
            qws[((size_t)b * SEQ + s) * EDIM + i]  = aq[j][r] + biq[r];  // Q: [S,E]
            kws[((size_t)b * EDIM + i) * SEQ + s]  = ak[j][r] + bik[r];  // K: [E,S]
            vws[((size_t)b * SEQ + s) * EDIM + i]  = av[j][r] + biv[r];  // V: [S,E]
        }
    }
}

// ---------------------------------------------------------------------------
// Kernel 2: streaming (flash-style) attention.
// Block = 128 threads (4 waves); each wave owns a 16-row query tile and the
// full E=128 output width in registers. Waves share LDS-staged K/V key tiles.
// While tile t is being consumed, tile t+1 is prefetched (global_prefetch_b8)
// so next iteration's cooperative loads hit cache.
// ---------------------------------------------------------------------------
__global__ __launch_bounds__(128) void attn_kernel(
    const float* __restrict__ qws, const float* __restrict__ kws,
    const float* __restrict__ vws, float* __restrict__ out)
{
    __shared__ float sK[EDIM][KTILE + 1];      // K^T tile: [e][t]
    __shared__ float sV[KTILE][EDIM + 1];      // V   tile: [t][e]
    __shared__ float sP[4][16][KTILE + 1];     // per-wave P tile (C->A bounce)

    const int tid  = threadIdx.x;
    const int lane = tid & 31;
    const int wave = tid >> 5;
    const int h    = lane >> 4;
    const int ln   = lane & 15;
    const int b    = blockIdx.y;
    const int s0   = blockIdx.x * 64 + wave * 16;

    // Cache this wave's Q tile as 32 ready-to-issue A fragments (16x128 fp32).
    v2f qf[32];
    {
        const float* qrow = qws + ((size_t)b * SEQ + s0 + ln) * EDIM;
#pragma unroll
        for (int ks = 0; ks < 32; ++ks)
            qf[ks] = *(const v2f*)(qrow + ks * 4 + h * 2);
    }

    v8f o[8];
#pragma unroll
    for (int et = 0; et < 8; ++et) o[et] = (v8f){};
    float m[8], l[8];
#pragma unroll
    for (int r = 0; r < 8; ++r) { m[r] = -1e30f; l[r] = 0.0f; }

    const float scale = 0.088388347648318447f;   // 1/sqrt(128)

    for (int t0 = 0; t0 < SEQ; t0 += KTILE) {
        // ---- cooperative stage of K^T and V tiles into LDS ----
        for (int idx = tid; idx < EDIM * KTILE; idx += 128) {
            const int e = idx >> 4, j = idx & 15;
            sK[e][j] = kws[((size_t)b * EDIM + e) * SEQ + t0 + j];
        }
        for (int idx = tid; idx < KTILE * EDIM; idx += 128) {
            const int j = idx >> 7, e = idx & 127;
            sV[j][e] = vws[((size_t)b * SEQ + t0 + j) * EDIM + e];
        }

        // ---- prefetch next key tile while this one is consumed ----
        if (t0 + KTILE < SEQ) {             // block-uniform branch: no lane divergence
            const int tn = t0 + KTILE;
            // next K tile: 128 rows of 64B, one per thread
            __builtin_prefetch(kws + ((size_t)b * EDIM + tid) * SEQ + tn, 0, 3);
            // next V tile: 16 rows x 512B, in 128B chunks (64 threads)
            if (tid < 64) {
                const int j = tid >> 2, c = tid & 3;
                __builtin_prefetch(vws + ((size_t)b * SEQ + tn + j) * EDIM + c * 32, 0, 3);
            }
        }
        __syncthreads();

        // ---- scores S = Q * K^T (16x16 tile, fp32 accum) ----
        v8f sc = (v8f){};
#pragma unroll
        for (int ks = 0; ks < 32; ++ks) {
            const int kb = ks * 4 + h * 2;
            v2f bf;
            bf.x = sK[kb][ln];
            bf.y = sK[kb + 1][ln];
            sc = wmma_f32(qf[ks], bf, sc);
        }

        // ---- online softmax update ----
        float rowm[8];
#pragma unroll
        for (int r = 0; r < 8; ++r) { sc[r] *= scale; rowm[r] = sc[r]; }
#pragma unroll
        for (int msk = 1; msk < 16; msk <<= 1)
#pragma unroll
            for (int r = 0; r < 8; ++r)
                rowm[r] = fmaxf(rowm[r], __shfl_xor(rowm[r], msk, 32));

        float corr[8];
#pragma unroll
        for (int r = 0; r < 8; ++r) {
            const float mn = fmaxf(m[r], rowm[r]);
            corr[r] = __expf(m[r] - mn);
            m[r] = mn;
            sc[r] = __expf(sc[r] - mn);
        }
        float rs[8];
#pragma unroll
        for (int r = 0; r < 8; ++r) rs[r] = sc[r];
#pragma unroll
        for (int msk = 1; msk < 16; msk <<= 1)
#pragma unroll
            for (int r = 0; r < 8; ++r)
                rs[r] += __shfl_xor(rs[r], msk, 32);
#pragma unroll
        for (int r = 0; r < 8; ++r) l[r] = l[r] * corr[r] + rs[r];
#pragma unroll
        for (int et = 0; et < 8; ++et)
#pragma unroll
            for (int r = 0; r < 8; ++r) o[et][r] *= corr[r];

        // ---- bounce P through LDS: C-layout -> A-layout ----
#pragma unroll
        for (int r = 0; r < 8; ++r)
            sP[wave][r + 8 * h][ln] = sc[r];

        // ---- O += P * V ----
#pragma unroll
        for (int ks = 0; ks < 4; ++ks) {
            const int kb = ks * 4 + h * 2;
            v2f pa;
            pa.x = sP[wave][ln][kb];
            pa.y = sP[wave][ln][kb + 1];
#pragma unroll
            for (int et = 0; et < 8; ++et) {
                v2f bf;
                bf.x = sV[kb][et * 16 + ln];
                bf.y = sV[kb + 1][et * 16 + ln];
                o[et] = wmma_f32(pa, bf, o[et]);
            }
        }
        __syncthreads();   // protect sK/sV for next tile
    }

    // ---- normalize and write out[b, s, e] ----
    float inv[8];
#pragma unroll
    for (int r = 0; r < 8; ++r) inv[r] = 1.0f / l[r];
#pragma unroll
    for (int et = 0; et < 8; ++et) {
        const int e = et * 16 + ln;
#pragma unroll
        for (int r = 0; r < 8; ++r) {
            const int s = s0 + r + 8 * h;
            out[((size_t)b * SEQ + s) * EDIM + e] = o[et][r] * inv[r];
        }
    }
}

// ---------------------------------------------------------------------------
extern "C" void kernel_launch(void* const* d_in, const int* in_sizes, int n_in,
                              void* d_out, int out_size, void* d_ws, size_t ws_size,
                              hipStream_t stream) {
    const float* x  = (const float*)d_in[0];
    const float* Wq = (const float*)d_in[1];
    const float* bq = (const float*)d_in[2];
    const float* Wk = (const float*)d_in[3];
    const float* bk = (const float*)d_in[4];
    const float* Wv = (const float*)d_in[5];
    const float* bv = (const float*)d_in[6];
    float* out = (float*)d_out;

    const size_t per = (size_t)BATCH * SEQ * EDIM;   // 2M floats = 8 MB each
    float* qws = (float*)d_ws;
    float* kws = qws + per;
    float* vws = kws + per;

    dim3 g1(SEQ / 64, EDIM / 64, BATCH);     // 64 x 2 x 4 blocks
    qkv_proj_kernel<<<g1, 128, 0, stream>>>(x, Wq, bq, Wk, bk, Wv, bv,
                                            qws, kws, vws);

    dim3 g2(SEQ / 64, BATCH);                // 64 x 4 blocks
    attn_kernel<<<g2, 128, 0, stream>>>(qws, kws, vws, out);
}